// MultiModalCausalUncertaintyNetwork_3384434230094
// MI455X (gfx1250) — compile-verified
//
#include <hip/hip_runtime.h>
#include <hip/hip_bf16.h>

// ---------------------------------------------------------------------------
// MI455X / gfx1250 implementation. All matmul-shaped stages use
// V_WMMA_F32_16X16X4_F32 (fp32-precise WMMA); attention is flash-style.
// ---------------------------------------------------------------------------

typedef __attribute__((ext_vector_type(2))) float v2f;
typedef __attribute__((ext_vector_type(8))) float v8f;

#define WMMA_F32(a, b, c) \
  __builtin_amdgcn_wmma_f32_16x16x4_f32(false, (a), false, (b), (short)0, (c), false, false)

// Model dims
#define BB 2
#define SS 512
#define EE 256
#define MM 4
#define HH 8
#define HD 32
#define LL (SS * MM)     // 2048
#define TT 4             // hist length (1 + 3 graph layers)

__constant__ int c_PI[12] = {0,0,0,1,1,1,2,2,2,3,3,3};
__constant__ int c_PJ[12] = {1,2,3,0,2,3,0,1,3,0,1,2};

__device__ __forceinline__ float sigmoidf_(float x) { return 1.0f / (1.0f + __expf(-x)); }

// ---------------------------------------------------------------------------
// Tiled GEMM: C[r,c] = act( sum_k A[r,k]*B[k,c] + bias[c] (+rowbias[r&mask,c]) )
// One wave computes a 16x32 C tile (two WMMA accumulators); 4 waves/block.
// All mode selection is compile-time; inner loop is branch-free:
//   1x global_load_b64 (A frag) + 4x B elements + 2x v_wmma per K-step.
// Requirements: rows % 16 == 0, N % 32 == 0, K % 16 == 0.
// GATHER: A row r = concat(attended[r/12, PI[r%12]], attended[r/12, PJ[r%12]])
// ---------------------------------------------------------------------------
template<int ACT, bool GATHER, bool RB>
__global__ __launch_bounds__(128)
void wmma_gemm_kernel(const float* __restrict__ A, long lda,
                      const float* __restrict__ Bm, long ldb,
                      const float* __restrict__ bias,
                      const float* __restrict__ rowbias, int rbMask, long rbLd,
                      float* __restrict__ C, long ldc,
                      int K, int N)
{
  const int wave = threadIdx.x >> 5;
  const int lane = threadIdx.x & 31;
  const int l16 = lane & 15, half = lane >> 4;
  const int row0 = blockIdx.x * 16;
  const int col0 = (blockIdx.y * 4 + wave) * 32;
  if (col0 >= N) return;
  const int arow = row0 + l16;

  v8f acc0 = {}, acc1 = {};
  const float* bp = Bm + (long)(2 * half) * ldb + col0 + l16;
  const long bstep = 4 * (long)ldb;

  if constexpr (GATHER) {
    const int pbs = arow / 12;
    const int pp  = arow % 12;
    const float* srcI = A + ((long)pbs * MM + c_PI[pp]) * EE + 2 * half;
    const float* srcJ = A + ((long)pbs * MM + c_PJ[pp]) * EE + 2 * half;
    #pragma unroll 1
    for (int hsel = 0; hsel < 2; ++hsel) {
      const float* ap = hsel ? srcJ : srcI;
      #pragma unroll 4
      for (int k0 = 0; k0 < EE; k0 += 4) {
        const v2f a = *(const v2f*)ap;
        v2f b0; b0.x = bp[0];  b0.y = bp[ldb];
        v2f b1; b1.x = bp[16]; b1.y = bp[ldb + 16];
        acc0 = WMMA_F32(a, b0, acc0);
        acc1 = WMMA_F32(a, b1, acc1);
        ap += 4; bp += bstep;
      }
    }
  } else {
    const float* ap = A + (long)arow * lda + 2 * half;
    #pragma unroll 4
    for (int k0 = 0; k0 < K; k0 += 4) {
      __builtin_prefetch(bp + 8 * bstep, 0, 1);   // speculative; safe past end
      const v2f a = *(const v2f*)ap;
      v2f b0; b0.x = bp[0];  b0.y = bp[ldb];
      v2f b1; b1.x = bp[16]; b1.y = bp[ldb + 16];
      acc0 = WMMA_F32(a, b0, acc0);
      acc1 = WMMA_F32(a, b1, acc1);
      ap += 4; bp += bstep;
    }
  }

  #pragma unroll
  for (int p = 0; p < 8; ++p) {
    const int r = row0 + p + 8 * half;
    #pragma unroll
    for (int n = 0; n < 2; ++n) {
      const int c = col0 + n * 16 + l16;
      float v = (n ? acc1[p] : acc0[p]) + bias[c];
      if constexpr (RB) v += rowbias[(long)(r & rbMask) * rbLd + c];
      if constexpr (ACT == 1) v = fmaxf(v, 0.0f);
      if constexpr (ACT == 2) v = sigmoidf_(v);
      C[(long)r * ldc + c] = v;
    }
  }
}

// ---------------------------------------------------------------------------
// Flash attention: one wave per (b,h, 16-row q tile). hd=32, causal.
// Scores via 8 WMMA steps; P redistributed through LDS; P@V via 8 WMMAs.
// ---------------------------------------------------------------------------
__global__ __launch_bounds__(32)
void flash_attn_kernel(const float* __restrict__ qkv, float* __restrict__ o)
{
  const int bh = blockIdx.x;
  const int b = bh / HH, h = bh % HH;
  const int q0 = blockIdx.y * 16;
  const int lane = threadIdx.x;
  const int l16 = lane & 15, half = lane >> 4;
  const long pitch = 3 * EE; // 768 floats per qkv row

  const float* qb = qkv + (long)b * LL * pitch + h * HD;
  const float* kb = qb + EE;
  const float* vb = qb + 2 * EE;
  const float scale = 0.17677669529663687f; // 1/sqrt(32)

  // Q tile as A-fragments (pre-scaled)
  v2f qa[8];
  const float* qrow = qb + (long)(q0 + l16) * pitch;
  #pragma unroll
  for (int ks = 0; ks < 8; ++ks) {
    const int ka = ks * 4 + 2 * half;
    qa[ks].x = qrow[ka] * scale;
    qa[ks].y = qrow[ka + 1] * scale;
  }

  v8f o0 = {}, o1 = {};
  float mrow[8], lrow[8];
  #pragma unroll
  for (int p = 0; p < 8; ++p) { mrow[p] = -3.0e38f; lrow[p] = 0.0f; }

  __shared__ float pt[16][17];

  for (int j0 = 0; j0 <= q0; j0 += 16) {
    // S = Q K^T (16x16 tile)
    v8f s = {};
    const float* krow = kb + (long)(j0 + l16) * pitch;
    #pragma unroll
    for (int ks = 0; ks < 8; ++ks) {
      const int ka = ks * 4 + 2 * half;
      v2f bfrag; bfrag.x = krow[ka]; bfrag.y = krow[ka + 1];
      s = WMMA_F32(qa[ks], bfrag, s);
    }

    const bool diag = (j0 == q0);
    #pragma unroll
    for (int p = 0; p < 8; ++p) {
      const int r = q0 + p + 8 * half;
      const int c = j0 + l16;
      float sv = s[p];
      if (diag && c > r) sv = -3.0e38f;
      // row max over the 16-lane group
      float mx = sv;
      #pragma unroll
      for (int d = 8; d > 0; d >>= 1) mx = fmaxf(mx, __shfl_xor(mx, d, 16));
      const float mnew = fmaxf(mrow[p], mx);
      const float corr = __expf(mrow[p] - mnew);
      const float pv   = __expf(sv - mnew);
      float rs = pv;
      #pragma unroll
      for (int d = 8; d > 0; d >>= 1) rs += __shfl_xor(rs, d, 16);
      lrow[p] = lrow[p] * corr + rs;
      mrow[p] = mnew;
      o0[p] *= corr; o1[p] *= corr;
      s[p] = pv;
    }

    // C-layout -> LDS -> A-layout for P
    #pragma unroll
    for (int p = 0; p < 8; ++p) pt[p + 8 * half][l16] = s[p];
    __syncthreads();
    v2f pa[4];
    #pragma unroll
    for (int ks = 0; ks < 4; ++ks) {
      const int ka = ks * 4 + 2 * half;
      pa[ks].x = pt[l16][ka]; pa[ks].y = pt[l16][ka + 1];
    }
    __syncthreads();

    // O += P V  (16x16 @ 16x32, two 16-col tiles)
    #pragma unroll
    for (int ks = 0; ks < 4; ++ks) {
      const int kk = j0 + ks * 4 + 2 * half;
      const float* vrow = vb + (long)kk * pitch;
      v2f b0; b0.x = vrow[l16];       b0.y = vrow[pitch + l16];
      v2f b1; b1.x = vrow[16 + l16];  b1.y = vrow[pitch + 16 + l16];
      o0 = WMMA_F32(pa[ks], b0, o0);
      o1 = WMMA_F32(pa[ks], b1, o1);
    }
  }

  #pragma unroll
  for (int p = 0; p < 8; ++p) {
    const int r = q0 + p + 8 * half;
    const float inv = 1.0f / lrow[p];
    float* orow = o + ((long)b * LL + r) * EE + h * HD;
    orow[l16]      = o0[p] * inv;
    orow[16 + l16] = o1[p] * inv;
  }
}

// ---------------------------------------------------------------------------
// Small GEMV-style linear for tiny tail stages: one thread per (r,c)
// ---------------------------------------------------------------------------
__global__ void gemv_kernel(const float* __restrict__ A, long lda,
                            const float* __restrict__ W,
                            const float* __restrict__ bias,
                            float* __restrict__ C, long ldc,
                            int rows, int K, int N, int act)
{
  const int t = blockIdx.x * blockDim.x + threadIdx.x;
  if (t >= rows * N) return;
  const int r = t / N, c = t % N;
  const float* a = A + (long)r * lda;
  float acc = bias[c];
  for (int k = 0; k < K; ++k) acc += a[k] * W[(long)k * N + c];
  if (act == 1)      acc = fmaxf(acc, 0.0f);
  else if (act == 2) acc = sigmoidf_(acc);
  C[(long)r * ldc + c] = acc;
}

// ---------------------------------------------------------------------------
// Elementwise / gather helpers for the graph tail
// ---------------------------------------------------------------------------
__global__ void init_xg_kernel(const float* __restrict__ attended,
                               const float* __restrict__ emb,
                               float* __restrict__ xg)
{
  const int idx = blockIdx.x * blockDim.x + threadIdx.x;
  if (idx >= BB * MM * EE) return;
  const int e = idx % EE, i = (idx / EE) % MM, b = idx / (MM * EE);
  xg[idx] = attended[(((long)b * SS + (SS - 1)) * MM + i) * EE + e] + emb[i * EE + e];
}

// graph edges (src,dst): (0,1),(0,2),(1,3),(2,3)
__constant__ int c_ESRC[4] = {0,0,1,2};
__constant__ int c_EDST[4] = {1,2,3,3};

__global__ void gather_ei_kernel(const float* __restrict__ xg, float* __restrict__ ei)
{
  const int idx = blockIdx.x * blockDim.x + threadIdx.x;
  if (idx >= 8 * 2 * EE) return;
  const int k = idx % (2 * EE);
  const int row = idx / (2 * EE);
  const int b = row / 4, eidx = row % 4;
  const int i = c_ESRC[eidx], nb = c_EDST[eidx];
  ei[idx] = (k < EE) ? xg[(b * MM + i) * EE + k] : xg[(b * MM + nb) * EE + (k - EE)];
}

__global__ void agg_ni_kernel(const float* __restrict__ xg,
                              const float* __restrict__ ew,
                              float* __restrict__ ni)
{
  const int idx = blockIdx.x * blockDim.x + threadIdx.x;
  if (idx >= BB * MM * EE) return;
  const int e = idx % EE, i = (idx / EE) % MM, b = idx / (MM * EE);
  float agg;
  if (i == 0)      agg = ew[b * 4 + 0] * xg[(b * MM + 1) * EE + e]
                       + ew[b * 4 + 1] * xg[(b * MM + 2) * EE + e];
  else if (i == 1) agg = ew[b * 4 + 2] * xg[(b * MM + 3) * EE + e];
  else if (i == 2) agg = ew[b * 4 + 3] * xg[(b * MM + 3) * EE + e];
  else             agg = xg[(b * MM + 3) * EE + e];
  ni[((long)(b * MM + i)) * (2 * EE) + e]      = xg[idx];
  ni[((long)(b * MM + i)) * (2 * EE) + EE + e] = agg;
}

// seq layout matches reference: stack(hist,axis=1).reshape(B*M,T,E), i.e.
// flat index ((b*T + t)*M + m)*E + e (T == M == 4 here).
__global__ void store_hist_kernel(const float* __restrict__ xg,
                                  float* __restrict__ seq, int t)
{
  const int idx = blockIdx.x * blockDim.x + threadIdx.x;
  if (idx >= BB * MM * EE) return;
  const int e = idx % EE, m = (idx / EE) % MM, b = idx / (MM * EE);
  seq[(((long)b * TT + t) * MM + m) * EE + e] = xg[idx];
}

__global__ void copy_kernel(const float* __restrict__ src, float* __restrict__ dst, int n)
{
  const int idx = blockIdx.x * blockDim.x + threadIdx.x;
  if (idx < n) dst[idx] = src[idx];
}

// ---------------------------------------------------------------------------
// One LSTM scan step: grid = 8 rows, block = 256 (one thread per feature)
// ---------------------------------------------------------------------------
__global__ __launch_bounds__(256)
void lstm_step_kernel(const float* __restrict__ xseq,
                      const float* __restrict__ Wih, const float* __restrict__ Whh,
                      const float* __restrict__ bih, const float* __restrict__ bhh,
                      float* __restrict__ hbuf, float* __restrict__ cbuf,
                      float* __restrict__ hout, int t)
{
  __shared__ float xs[EE], hs[EE];
  const int r = blockIdx.x;
  const int e = threadIdx.x;
  xs[e] = xseq[((long)r * TT + t) * EE + e];
  hs[e] = (t == 0) ? 0.0f : hbuf[r * EE + e];
  __syncthreads();

  float g[4];
  #pragma unroll
  for (int gi = 0; gi < 4; ++gi) {
    const int grow = gi * EE + e;
    const float* wi = Wih + (long)grow * EE;
    const float* wh = Whh + (long)grow * EE;
    float acc = bih[grow] + bhh[grow];
    for (int k = 0; k < EE; ++k) acc += wi[k] * xs[k] + wh[k] * hs[k];
    g[gi] = acc;
  }
  const float ig = sigmoidf_(g[0]);
  const float fg = sigmoidf_(g[1]);
  const float gg = tanhf(g[2]);
  const float og = sigmoidf_(g[3]);
  float c = (t == 0) ? 0.0f : cbuf[r * EE + e];
  c = fg * c + ig * gg;
  const float h = og * tanhf(c);
  cbuf[r * EE + e] = c;
  hbuf[r * EE + e] = h;
  hout[((long)r * TT + t) * EE + e] = h;
}

// ---------------------------------------------------------------------------
// Host-side launcher
// ---------------------------------------------------------------------------
static inline void run_gemm(hipStream_t s, const float* A, long lda,
                            const float* B, long ldb, const float* bias,
                            const float* rowbias, int rbMask, long rbLd,
                            float* C, long ldc, int rows, int K, int N,
                            int act, int gather)
{
  dim3 grid(rows / 16, (N + 127) / 128);
  if (gather) {
    wmma_gemm_kernel<1, true, false><<<grid, 128, 0, s>>>(
        A, lda, B, ldb, bias, nullptr, 0, 0, C, ldc, K, N);
  } else if (rowbias) {
    wmma_gemm_kernel<0, false, true><<<grid, 128, 0, s>>>(
        A, lda, B, ldb, bias, rowbias, rbMask, rbLd, C, ldc, K, N);
  } else if (act == 1) {
    wmma_gemm_kernel<1, false, false><<<grid, 128, 0, s>>>(
        A, lda, B, ldb, bias, nullptr, 0, 0, C, ldc, K, N);
  } else {
    wmma_gemm_kernel<0, false, false><<<grid, 128, 0, s>>>(
        A, lda, B, ldb, bias, nullptr, 0, 0, C, ldc, K, N);
  }
}

static inline void run_gemv(hipStream_t s, const float* A, long lda,
                            const float* W, const float* bias, float* C, long ldc,
                            int rows, int K, int N, int act)
{
  const int tot = rows * N;
  gemv_kernel<<<(tot + 255) / 256, 256, 0, s>>>(A, lda, W, bias, C, ldc, rows, K, N, act);
}

extern "C" void kernel_launch(void* const* d_in, const int* in_sizes, int n_in,
                              void* d_out, int out_size, void* d_ws, size_t ws_size,
                              hipStream_t stream)
{
  (void)in_sizes; (void)n_in; (void)out_size; (void)ws_size;
  #define IN(i) ((const float*)d_in[(i)])
  // --- input leaf indices (JAX sorted-pytree order: params then x) ---
  const float* attn_Wo   = IN(0);   const float* attn_Wqkv = IN(1);
  const float* attn_bo   = IN(2);   const float* attn_bqkv = IN(3);
  const float* attn_pos  = IN(4);
  const float* fus_W1 = IN(5); const float* fus_W2 = IN(6);
  const float* fus_b1 = IN(7); const float* fus_b2 = IN(8);
  const float* edge_W1 = IN(9);  const float* edge_W2 = IN(10);
  const float* edge_b1 = IN(11); const float* edge_b2 = IN(12);
  const float* g_emb = IN(13);
  const float* lyr_W1[3] = {IN(14), IN(18), IN(22)};
  const float* lyr_W2[3] = {IN(15), IN(19), IN(23)};
  const float* lyr_b1[3] = {IN(16), IN(20), IN(24)};
  const float* lyr_b2[3] = {IN(17), IN(21), IN(25)};
  const float* lstm_Whh[2] = {IN(26), IN(30)};
  const float* lstm_Wih[2] = {IN(27), IN(31)};
  const float* lstm_bhh[2] = {IN(28), IN(32)};
  const float* lstm_bih[2] = {IN(29), IN(33)};
  const float* ip_W = IN(34); const float* ip_b = IN(35);
  const float* me_W1[4] = {IN(36), IN(40), IN(44), IN(48)};
  const float* me_W2[4] = {IN(37), IN(41), IN(45), IN(49)};
  const float* me_b1[4] = {IN(38), IN(42), IN(46), IN(50)};
  const float* me_b2[4] = {IN(39), IN(43), IN(47), IN(51)};
  const float* mp_W[4] = {IN(52), IN(54), IN(56), IN(58)};
  const float* mp_b[4] = {IN(53), IN(55), IN(57), IN(59)};
  const float* pr_W1[4] = {IN(60), IN(64), IN(68), IN(72)};
  const float* pr_W2[4] = {IN(61), IN(65), IN(69), IN(73)};
  const float* pr_b1[4] = {IN(62), IN(66), IN(70), IN(74)};
  const float* pr_b2[4] = {IN(63), IN(67), IN(71), IN(75)};
  const float* st_W1 = IN(76); const float* st_W2 = IN(77);
  const float* st_b1 = IN(78); const float* st_b2 = IN(79);
  const float* x = IN(80);

  // --- output slices ---
  float* out = (float*)d_out;
  float* out_attended = out;                          // B*S*M*E = 1048576
  float* out_str      = out + 1048576;                // B*S*12  = 12288
  float* out_xg       = out + 1060864;                // B*M*E   = 2048
  float* out_dyn      = out + 1062912;                // B*M*T*E = 8192
  float* out_unc      = out + 1071104;                // B*M*2   = 16
  float* out_fused    = out + 1071120;                // B*2     = 4

  // --- workspace layout (floats) ---
  float* ws = (float*)d_ws;
  float* xp   = ws;                 // 1024 x 256
  float* u    = ws + 262144;        // 4096 x 256 (modes+pos; reused as attn-out)
  float* qkv  = ws + 1310720;       // 4096 x 768
  float* obuf = ws + 4456448;       // 4096 x 256
  float* hbuf = ws + 5505024;       // up to 12288 x 256 hidden scratch
  float* SBp  = ws + 8650752;       // small area
  float* xg   = SBp + 0;            // 2048
  float* ei   = SBp + 8192;         // 8 x 512
  float* eh   = SBp + 16384;        // 8 x 256
  float* ew   = SBp + 24576;        // 8
  float* ni   = SBp + 32768;        // 8 x 512
  float* nxg  = SBp + 40960;        // 2048
  float* seq0 = SBp + 49152;        // 8 x 4 x 256
  float* seq1 = SBp + 57344;        // 8 x 4 x 256
  float* lh   = SBp + 65536;        // 8 x 256 lstm h
  float* lc   = SBp + 73728;        // 8 x 256 lstm c
  float* fh   = SBp + 81920;        // fusion hidden 2 x 256
  float* ph   = SBp + 90112;        // pred hidden 2 x 128

  const int rowsBS = BB * SS;   // 1024
  const int rowsBL = BB * LL;   // 4096

  // 1. xp = x @ Wip + bip
  run_gemm(stream, x, EE, ip_W, EE, ip_b, nullptr, 0, 0, xp, EE, rowsBS, EE, EE, 0, 0);

  // 2. mode encoders: u[:, :, m] = mlp2(xp) + pos[s]
  for (int m = 0; m < MM; ++m) {
    run_gemm(stream, xp, EE, me_W1[m], EE, me_b1[m], nullptr, 0, 0,
             hbuf, EE, rowsBS, EE, EE, 1, 0);
    run_gemm(stream, hbuf, EE, me_W2[m], EE, me_b2[m], attn_pos, SS - 1, EE,
             u + m * EE, MM * EE, rowsBS, EE, EE, 0, 0);
  }

  // 3. qkv = u @ Wqkv + bqkv   (u viewed as (4096,256))
  run_gemm(stream, u, EE, attn_Wqkv, 3 * EE, attn_bqkv, nullptr, 0, 0,
           qkv, 3 * EE, rowsBL, EE, 3 * EE, 0, 0);

  // 4. flash attention -> obuf (B*L, E)
  {
    dim3 grid(BB * HH, LL / 16);
    flash_attn_kernel<<<grid, 32, 0, stream>>>(qkv, obuf);
  }

  // 5. attn output projection (reuse u): u = obuf @ Wo + bo
  run_gemm(stream, obuf, EE, attn_Wo, EE, attn_bo, nullptr, 0, 0,
           u, EE, rowsBL, EE, EE, 0, 0);

  // 6. per-mode projections -> attended (directly into d_out)
  for (int m = 0; m < MM; ++m) {
    run_gemm(stream, u + m * EE, MM * EE, mp_W[m], EE, mp_b[m], nullptr, 0, 0,
             out_attended + m * EE, MM * EE, rowsBS, EE, EE, 0, 0);
  }

  // 7. pairwise strengths: gather-GEMM (12288 x 512 x 256, relu) + gemv sigmoid
  run_gemm(stream, out_attended, 0, st_W1, EE, st_b1, nullptr, 0, 0,
           hbuf, EE, rowsBS * 12, 2 * EE, EE, 1, /*gather=*/1);
  run_gemv(stream, hbuf, EE, st_W2, st_b2, out_str, 1, rowsBS * 12, EE, 1, 2);

  // 8. graph init: xg = attended[:, -1] + emb ; hist[0]
  init_xg_kernel<<<(BB*MM*EE + 255) / 256, 256, 0, stream>>>(out_attended, g_emb, xg);
  store_hist_kernel<<<(BB*MM*EE + 255) / 256, 256, 0, stream>>>(xg, seq0, 0);

  // 9. graph layers
  for (int l = 0; l < 3; ++l) {
    gather_ei_kernel<<<(8 * 2 * EE + 255) / 256, 256, 0, stream>>>(xg, ei);
    run_gemv(stream, ei, 2 * EE, edge_W1, edge_b1, eh, EE, 8, 2 * EE, EE, 1);
    run_gemv(stream, eh, EE, edge_W2, edge_b2, ew, 1, 8, EE, 1, 2);
    agg_ni_kernel<<<(BB*MM*EE + 255) / 256, 256, 0, stream>>>(xg, ew, ni);
    run_gemv(stream, ni, 2 * EE, lyr_W1[l], lyr_b1[l], eh, EE, 8, 2 * EE, EE, 1);
    run_gemv(stream, eh, EE, lyr_W2[l], lyr_b2[l], nxg, EE, 8, EE, EE, 0);
    copy_kernel<<<(BB*MM*EE + 255) / 256, 256, 0, stream>>>(nxg, xg, BB * MM * EE);
    store_hist_kernel<<<(BB*MM*EE + 255) / 256, 256, 0, stream>>>(xg, seq0, l + 1);
  }

  // 10. xg output
  copy_kernel<<<(BB*MM*EE + 255) / 256, 256, 0, stream>>>(xg, out_xg, BB * MM * EE);

  // 11. two LSTM layers over the 4-step hist sequence; layer 2 writes dyn
  for (int t = 0; t < TT; ++t)
    lstm_step_kernel<<<8, EE, 0, stream>>>(seq0, lstm_Wih[0], lstm_Whh[0],
                                           lstm_bih[0], lstm_bhh[0], lh, lc, seq1, t);
  for (int t = 0; t < TT; ++t)
    lstm_step_kernel<<<8, EE, 0, stream>>>(seq1, lstm_Wih[1], lstm_Whh[1],
                                           lstm_bih[1], lstm_bhh[1], lh, lc, out_dyn, t);

  // 12. per-mode uncertainty heads
  for (int i = 0; i < MM; ++i) {
    run_gemv(stream, xg + i * EE, MM * EE, pr_W1[i], pr_b1[i], ph, 128, 2, EE, 128, 1);
    run_gemv(stream, ph, 128, pr_W2[i], pr_b2[i], out_unc + i * 2, MM * 2, 2, 128, 2, 0);
  }

  // 13. fusion head
  run_gemv(stream, xg, MM * EE, fus_W1, fus_b1, fh, EE, 2, MM * EE, EE, 1);
  run_gemv(stream, fh, EE, fus_W2, fus_b2, out_fused, 2, 2, EE, 2, 0);
  #undef IN
}